// MeanPooling_38122129719300
// MI455X (gfx1250) — compile-verified
//
#include <hip/hip_runtime.h>
#include <hip/hip_bf16.h>
#include <stdint.h>

// Mean pooling over time with length mask:
//   out[b, d] = (1/len_b) * sum_{t < len_b} xs[b, t, d]
// B=32, T=4096, D=1024, fp32.  Bandwidth-bound (0.25 FLOP/byte):
// ~512MB / 23.3TB/s ~ 22us floor; chunk skipping halves average traffic.
// Strategy: async global->LDS staging (ASYNCcnt, double-buffered) +
// V_WMMA_F32_16X16X4_F32 masked reduction.  Mask compares are hoisted out
// of the hot loop: full chunks use a constant all-ones A fragment; only
// the single tail chunk computes per-element masks.

typedef __attribute__((ext_vector_type(2))) float v2f;
typedef __attribute__((ext_vector_type(8))) float v8f;

#define BB 32
#define TT 4096
#define DD 1024
#define TCHUNK 64
#define BUF_FLOATS (TCHUNK * 16) // 1024 floats = 4KB per buffer

struct FullMask { static constexpr bool value = true;  };
struct TailMask { static constexpr bool value = false; };

__device__ __forceinline__ void async_b128(uint32_t lds_off, const float* gp) {
    // GLOBAL_LOAD_ASYNC_TO_LDS_B128: per-lane 16B global -> LDS, ASYNCcnt-tracked
    asm volatile("global_load_async_to_lds_b128 %0, %1, off"
                 :: "v"(lds_off), "v"((unsigned long long)(uintptr_t)gp)
                 : "memory");
}

__global__ __launch_bounds__(32)
void MeanPooling_38122129719300_kernel(const float* __restrict__ xs,
                                       const int* __restrict__ xs_len,
                                       float* __restrict__ out) {
    __shared__ float tile[2 * BUF_FLOATS]; // double buffer, 8KB

    const int lane = threadIdx.x;          // wave32: one wave per block
    const int bidx = blockIdx.x;
    const int b    = bidx >> 6;            // D/16 = 64 tiles per batch
    const int d0   = (bidx & 63) << 4;

    const int len     = xs_len[b];                       // 1..T
    const int nfull   = len / TCHUNK;                    // all-valid chunks
    const int nchunks = (len + TCHUNK - 1) / TCHUNK;     // nfull or nfull+1

    // async-copy lane mapping: 8 rows x 4 colgroups per instruction
    const int lr = lane >> 2;   // row 0..7 within 8-row group
    const int lc = lane & 3;    // 4-float column group
    // WMMA lane mapping: lane-half h -> K = 2h + j ; n = N column
    const int half = lane >> 4;
    const int n    = lane & 15;

    const uint32_t lds_base = (uint32_t)(uintptr_t)(void*)tile;
    const float* gsrc = xs + (size_t)b * (size_t)TT * (size_t)DD + (size_t)d0;

    auto issue = [&](int ck) {
        // drain DS reads of the buffer being overwritten before async writes
        asm volatile("s_wait_dscnt 0x0" ::: "memory");
        const float* gp = gsrc + (size_t)(ck * TCHUNK + lr) * DD + lc * 4;
        const uint32_t loff = lds_base +
            (uint32_t)((ck & 1) * (BUF_FLOATS * 4) + (lr * 16 + lc * 4) * 4);
#pragma unroll
        for (int i = 0; i < 8; ++i)                 // 8 x 512B = 4KB chunk
            async_b128(loff + i * 512, gp + (size_t)i * 8 * DD);
    };

    v8f acc = {};
    const v2f ones = {1.0f, 1.0f};   // constant A fragment for full chunks

    auto compute = [&](int ck, auto FULLC) {
        constexpr bool FULL = decltype(FULLC)::value;
        const int base  = (ck & 1) * BUF_FLOATS;
        const int tglob = ck * TCHUNK;
#pragma unroll
        for (int tg = 0; tg < 16; ++tg) {          // 16 x (K=4) = 64 t per chunk
            const int tl = tg * 4 + 2 * half;      // this lane-half's K rows
            v2f a, bv;
            if constexpr (FULL) {
                a = ones;                          // no compares in hot loop
            } else {
                a.x = (tglob + tl     < len) ? 1.0f : 0.0f;
                a.y = (tglob + tl + 1 < len) ? 1.0f : 0.0f;
            }
            // B[k][n] = xs[t, d0+n] from LDS (merges into ds_load_2addr_b32)
            bv.x = tile[base + tl * 16 + n];
            bv.y = tile[base + (tl + 1) * 16 + n];
            // D = A(16x4) * B(4x16) + C : fp32 WMMA, accumulate over time
            acc = __builtin_amdgcn_wmma_f32_16x16x4_f32(
                false, a, false, bv, (short)0, acc, false, false);
        }
    };

    issue(0);
    for (int ck = 0; ck < nchunks; ++ck) {
        if (ck + 1 < nchunks) {
            issue(ck + 1);
            // in-order completion: <=8 outstanding means chunk ck has landed
            asm volatile("s_wait_asynccnt 0x8" ::: "memory");
        } else {
            asm volatile("s_wait_asynccnt 0x0" ::: "memory");
        }
        if (ck < nfull) compute(ck, FullMask{});   // uniform scalar branch
        else            compute(ck, TailMask{});   // (EXEC all-1s for WMMA)
    }

    // C/D layout: VGPR0, lanes 0-15 = row M=0, N=lane (all rows identical)
    if (lane < 16) {
        const float inv_len = 1.0f / (float)len;
        out[(size_t)b * DD + d0 + lane] = acc[0] * inv_len;
    }
}

extern "C" void kernel_launch(void* const* d_in, const int* in_sizes, int n_in,
                              void* d_out, int out_size, void* d_ws, size_t ws_size,
                              hipStream_t stream) {
    (void)in_sizes; (void)n_in; (void)out_size; (void)d_ws; (void)ws_size;
    const float* xs     = (const float*)d_in[0];
    const int*   xs_len = (const int*)d_in[1];
    float*       out    = (float*)d_out;

    dim3 grid(BB * (DD / 16));  // 2048 workgroups
    dim3 block(32);             // one wave32 each
    hipLaunchKernelGGL(MeanPooling_38122129719300_kernel, grid, block, 0, stream,
                       xs, xs_len, out);
}